// GCNEncoder_18614388261506
// MI455X (gfx1250) — compile-verified
//
#include <hip/hip_runtime.h>
#include <cmath>

#define CH 128            // IN_CH == HID_CH == 128
#define LDSW_STRIDE 136   // 128 + 8 floats pad: half-waves hit disjoint LDS bank sets

typedef __attribute__((ext_vector_type(2))) float v2f;
typedef __attribute__((ext_vector_type(8))) float v8f;
typedef __attribute__((ext_vector_type(4))) unsigned int u32x4;
typedef __attribute__((ext_vector_type(8))) int i32x8;
typedef __attribute__((ext_vector_type(4))) int i32x4;

// ---------------- degree / normalization ----------------

__global__ void deg_init_kernel(float* deg, int N) {
  int i = blockIdx.x * blockDim.x + threadIdx.x;
  if (i < N) deg[i] = 1.0f;  // self-loop contributes 1 to every node
}

__global__ void deg_count_kernel(const int* __restrict__ dst, float* deg, int E) {
  int e = blockIdx.x * blockDim.x + threadIdx.x;
  if (e < E) atomicAdd(&deg[dst[e]], 1.0f);
}

__global__ void dinv_kernel(float* deg, int N) {
  int i = blockIdx.x * blockDim.x + threadIdx.x;
  if (i < N) {
    float d = deg[i];
    deg[i] = (d > 0.0f) ? rsqrtf(d) : 0.0f;  // deg >= 1 always (self loops)
  }
}

// ---------------- dense GEMM: C[M x 128] = A[M x 128] * W[128 x 128] ----------------
// W staged into LDS once per block by the Tensor Data Mover (wave 0), with TDM
// padding of 8 DWORDs per 128-DWORD row (-> row stride 136 floats, bank-conflict free).
// One wave per 16x16 output tile via V_WMMA_F32_16X16X4_F32 (fp32 = reference precision).
// 32-bit A 16x4: VGPR v, lanes 0-15 -> K=v, lanes 16-31 -> K=v+2, row M=lane&15.
// 32-bit B 4x16: VGPR v, lanes 0-15 -> K=v, lanes 16-31 -> K=v+2, col N=lane&15.
// 32-bit C/D 16x16: VGPR r, lanes 0-15 -> M=r, lanes 16-31 -> M=r+8, N=lane&15.
__global__ void __launch_bounds__(256)
gemm_wmma_kernel(const float* __restrict__ A, const float* __restrict__ W,
                 float* __restrict__ C, int M) {
  __shared__ float sW[CH * LDSW_STRIDE];

#if __has_builtin(__builtin_amdgcn_tensor_load_to_lds) && __has_builtin(__builtin_amdgcn_s_wait_tensorcnt)
  if (threadIdx.x < 32) {  // wave 0 drives the TDM; EXEC fully set for this wave
    unsigned long long ga = (unsigned long long)(uintptr_t)W;
    u32x4 g0;
    g0[0] = 1u;                                   // count=1 valid user descriptor
    g0[1] = 0u;                                   // lds_addr = 0 (sW is the only LDS object)
    g0[2] = (unsigned int)(ga & 0xffffffffull);   // global_addr[31:0]
    g0[3] = (unsigned int)((ga >> 32) & 0x01ffffffull) | (2u << 30);  // addr[56:32] | type=2
    i32x8 g1;
    g1[0] = (2 << 16)                 // data_size = 4 bytes
          | (1 << 20)                 // pad_enable (LDS destination padding)
          | (6 << 22)                 // pad_interval code 6 = every 128 DWORDs
          | (7 << 25);                // pad_amount  code 7 = 8 DWORDs
    g1[1] = (CH & 0xffff) << 16;      // tensor_dim0[15:0] = 128
    g1[2] = (CH & 0xffff) << 16;      // tensor_dim1[15:0] = 128 (dim0 hi16 = 0)
    g1[3] = (CH & 0xffff) << 16;      // tile_dim0 = 128     (dim1 hi16 = 0)
    g1[4] = CH;                       // tile_dim1 = 128, tile_dim2 = 0
    g1[5] = CH;                       // tensor_dim0_stride[31:0] = 128
    g1[6] = 0;                        // stride hi / dim1_stride lo (unused, 2D)
    g1[7] = 0;
    i32x4 gz4 = {0, 0, 0, 0};             // groups 2/3 unused (2D tensor)
    i32x8 gz8 = {0, 0, 0, 0, 0, 0, 0, 0}; // trailing group (6-arg toolchain form)
    __builtin_amdgcn_tensor_load_to_lds(g0, g1, gz4, gz4, gz8, 0);
    __builtin_amdgcn_s_wait_tensorcnt(0);
  }
  // The TDM writes LDS behind the compiler's back: escape sW with a memory
  // clobber so the ds_loads below are not folded as reads of undef and the
  // LDS allocation is not eliminated.
  asm volatile("" : : "v"(&sW[0]) : "memory");
#else
  for (int i = threadIdx.x; i < CH * CH / 4; i += 256) {
    float4 v = ((const float4*)W)[i];
    int r = (i * 4) / CH, c = (i * 4) % CH;
    *(float4*)&sW[r * LDSW_STRIDE + c] = v;
  }
#endif
  __syncthreads();

  const int wave = threadIdx.x >> 5;   // 0..7 -> column tile
  const int lane = threadIdx.x & 31;
  const int half = lane >> 4;          // 0: lanes 0-15, 1: lanes 16-31
  const int l    = lane & 15;
  const int row0 = blockIdx.x * 16;    // grid is exact: M % 16 == 0
  const int col0 = wave * 16;

  const float* arow = A + (size_t)(row0 + l) * CH + half * 2;      // global (streamed)
  const float* brow = &sW[(half * 2) * LDSW_STRIDE + col0 + l];    // LDS

  v8f acc = {};
#pragma unroll 4
  for (int k = 0; k < CH; k += 4) {
    v2f a, b;
    a.x = arow[k];
    a.y = arow[k + 1];
    b.x = brow[(size_t)k * LDSW_STRIDE];
    b.y = brow[(size_t)(k + 1) * LDSW_STRIDE];
    acc = __builtin_amdgcn_wmma_f32_16x16x4_f32(
        /*neg_a=*/false, a, /*neg_b=*/false, b,
        /*c_mod=*/(short)0, acc, /*reuse_a=*/false, /*reuse_b=*/false);
  }

  float* cptr = C + (size_t)(row0 + half * 8) * CH + col0 + l;
#pragma unroll
  for (int r = 0; r < 8; ++r)
    cptr[(size_t)r * CH] = acc[r];
}

// ---------------- aggregation ----------------

// agg[i,c] = h[i,c] * dinv[i]^2 (self loop) + bias[c]; one float4 per thread
__global__ void agg_init_kernel(const float* __restrict__ h, const float* __restrict__ dinv,
                                const float* __restrict__ bias, float* __restrict__ agg,
                                long long total4) {  // total4 = N * 32
  long long t = (long long)blockIdx.x * blockDim.x + threadIdx.x;
  if (t >= total4) return;
  int i = (int)(t >> 5);         // node
  int q = (int)(t & 31);         // float4 index within the 128-ch row
  float di = dinv[i];
  float w = di * di;
  float4 hv = ((const float4*)h)[t];
  float4 bv = ((const float4*)bias)[q];
  float4 o;
  o.x = hv.x * w + bv.x;
  o.y = hv.y * w + bv.y;
  o.z = hv.z * w + bv.z;
  o.w = hv.w * w + bv.w;
  ((float4*)agg)[t] = o;
}

// one WAVE per edge: scalar (SGPR) edge metadata, one float4 gather + 4 atomics per lane
__global__ void edge_scatter_kernel(const float* __restrict__ h, const float* __restrict__ dinv,
                                    const int* __restrict__ src, const int* __restrict__ dst,
                                    float* __restrict__ agg, int E) {
  int wid = blockIdx.x * (blockDim.x >> 5) + (threadIdx.x >> 5);
  if (wid >= E) return;
  int e = __builtin_amdgcn_readfirstlane(wid);  // provably wave-uniform -> scalar loads
  int lane = threadIdx.x & 31;
  int s = src[e];
  int d = dst[e];
  float w = dinv[s] * dinv[d];
  float4 v = ((const float4*)(h + (size_t)s * CH))[lane];  // coalesced b128
  float* a = agg + (size_t)d * CH + lane * 4;
  atomicAdd(a + 0, v.x * w);
  atomicAdd(a + 1, v.y * w);
  atomicAdd(a + 2, v.z * w);
  atomicAdd(a + 3, v.w * w);
}

__global__ void tanh_kernel(float* __restrict__ out, long long total) {
  long long t = (long long)blockIdx.x * blockDim.x + threadIdx.x;
  if (t < total) out[t] = tanhf(out[t]);
}

// ---------------- host orchestration ----------------

extern "C" void kernel_launch(void* const* d_in, const int* in_sizes, int n_in,
                              void* d_out, int out_size, void* d_ws, size_t ws_size,
                              hipStream_t stream) {
  const float* x  = (const float*)d_in[0];
  const int*   ei = (const int*)d_in[1];   // [2, E] row-major: src row then dst row
  const float* W1 = (const float*)d_in[2];
  const float* b1 = (const float*)d_in[3];
  const float* W2 = (const float*)d_in[4];
  const float* b2 = (const float*)d_in[5];

  const int N = in_sizes[0] / CH;   // 50000
  const int E = in_sizes[1] / 2;    // 800000
  const int* src = ei;
  const int* dst = ei + E;

  // workspace: dinv | h (N*CH) | agg1 (N*CH); h buffer reused for h2; agg2 lives in d_out
  char*  ws   = (char*)d_ws;
  float* dinv = (float*)ws;
  size_t off  = (((size_t)N * sizeof(float)) + 255) & ~(size_t)255;
  float* h    = (float*)(ws + off);
  off        += (size_t)N * CH * sizeof(float);
  float* agg1 = (float*)(ws + off);
  float* out  = (float*)d_out;

  const long long totalNC  = (long long)N * CH;
  const long long totalNC4 = totalNC >> 2;        // float4 count
  const int B = 256;
  const int wavesPerBlock = B / 32;
  const int gN   = (N + B - 1) / B;
  const int gE   = (E + B - 1) / B;
  const int gNC4 = (int)((totalNC4 + B - 1) / B);
  const int gNC  = (int)((totalNC + B - 1) / B);
  const int gEW  = (E + wavesPerBlock - 1) / wavesPerBlock;  // one wave per edge
  const int gMM  = N / 16;                                   // 50000 % 16 == 0

  // normalization: deg -> dinv (in place)
  deg_init_kernel<<<gN, B, 0, stream>>>(dinv, N);
  deg_count_kernel<<<gE, B, 0, stream>>>(dst, dinv, E);
  dinv_kernel<<<gN, B, 0, stream>>>(dinv, N);

  // layer 1: h1 = x @ W1 ; agg1 = self + b1 + scatter(norm * h1[src])
  gemm_wmma_kernel<<<gMM, B, 0, stream>>>(x, W1, h, N);
  agg_init_kernel<<<gNC4, B, 0, stream>>>(h, dinv, b1, agg1, totalNC4);
  edge_scatter_kernel<<<gEW, B, 0, stream>>>(h, dinv, src, dst, agg1, E);

  // layer 2: h2 = agg1 @ W2 (reuse h buffer) ; agg2 accumulated directly in d_out
  gemm_wmma_kernel<<<gMM, B, 0, stream>>>(agg1, W2, h, N);
  agg_init_kernel<<<gNC4, B, 0, stream>>>(h, dinv, b2, out, totalNC4);
  edge_scatter_kernel<<<gEW, B, 0, stream>>>(h, dinv, src, dst, out, E);

  // final activation in-place
  tanh_kernel<<<gNC, B, 0, stream>>>(out, totalNC);
}